// BNB4bitEmbedding_77068893159749
// MI455X (gfx1250) — compile-verified
//
#include <hip/hip_runtime.h>
#include <stdint.h>

// NF4 embedding dequant-gather for MI455X (gfx1250, wave32).
// Store-bandwidth bound: ~128 MiB out + ~16 MiB gathered reads -> ~6.2 us
// at 23.3 TB/s. No matrix contraction exists here, so WMMA is inapplicable;
// the CDNA5 lever is the wave32 lane-resident codebook via ds_bpermute_b32.
//
// v2 changes vs v1:
//  - 16 floats/thread (64 threads/token, 2 waves/token): halves fixed
//    per-thread overhead (uniform loads, tid math) per output byte.
//  - wave-interleaved chunk layout: every global_store_b128 writes 512B
//    fully contiguous per wave (full cachelines, minimal GL2 requests);
//    every u16 packed load reads 64B contiguous per wave.

__global__ __launch_bounds__(256) void nf4_embed_kernel(
    const int* __restrict__ x,              // [T] token ids
    const uint8_t* __restrict__ packed,     // [V, halfD]
    const float* __restrict__ absmax,       // [V]
    const float* __restrict__ codebook,     // [16]
    float* __restrict__ out,                // [T, D]
    int total_threads,                      // T * D / 16
    int tok_shift,                          // log2(D/16)
    int tok_mask,                           // (D/16) - 1
    int halfD,                              // D/2 bytes per packed row
    int D)
{
    const int t = blockIdx.x * blockDim.x + threadIdx.x;
    if (t >= total_threads) return;

    const int lane = threadIdx.x & 31;      // wave32 lane id

    // Lane-resident codebook: lanes 0..15 hold the table (16..31 mirror).
    const float cb = codebook[lane & 15];

    const int tok = t >> tok_shift;         // 64 threads per token (D=1024)
    const int sub = t & tok_mask;
    const int w   = sub >> 5;               // which wave-half of the token

    const int row = x[tok];                 // wave-uniform
    const float scale = absmax[row];        // wave-uniform

    // Pre-scale the codebook once per thread -> no per-element multiply.
    const int cbs = __float_as_int(cb * scale);

    // Wave covers 512 floats (2KB out) = 256 packed bytes.
    // Chunk k: lane l handles float4 at  w*512 + k*128 + l*4   (outputs)
    //          and u16 at               w*256 + k*64  + l*2   (packed bytes)
    const uint8_t* prow = packed + (size_t)row * (size_t)halfD
                        + ((size_t)w << 8) + ((size_t)lane << 1);
    float* obase = out + (size_t)tok * (size_t)D
                 + ((size_t)w << 9) + ((size_t)lane << 2);

    // Issue all packed loads up front; 64B dense per wave-instruction.
    uint32_t h[4];
#pragma unroll
    for (int k = 0; k < 4; ++k)
        h[k] = *(const uint16_t*)(prow + (k << 6));

#pragma unroll
    for (int k = 0; k < 4; ++k) {
        const uint32_t v = h[k];
        // byte0 -> outputs +0 (high nib), +1 (low nib); byte1 -> +2, +3
        const int n0 = (int)((v >> 4)  & 0xFu);
        const int n1 = (int)( v        & 0xFu);
        const int n2 = (int)((v >> 12) & 0xFu);
        const int n3 = (int)((v >> 8)  & 0xFu);
        // ds_bpermute: dst[lane] = src[index]; index in bytes (<<2).
        const int r0 = __builtin_amdgcn_ds_bpermute(n0 << 2, cbs);
        const int r1 = __builtin_amdgcn_ds_bpermute(n1 << 2, cbs);
        const int r2 = __builtin_amdgcn_ds_bpermute(n2 << 2, cbs);
        const int r3 = __builtin_amdgcn_ds_bpermute(n3 << 2, cbs);
        // 512B fully contiguous per wave per store instruction.
        *(float4*)(obase + (k << 7)) =
            make_float4(__int_as_float(r0), __int_as_float(r1),
                        __int_as_float(r2), __int_as_float(r3));
    }
}

extern "C" void kernel_launch(void* const* d_in, const int* in_sizes, int n_in,
                              void* d_out, int out_size, void* d_ws, size_t ws_size,
                              hipStream_t stream) {
    const int*      x        = (const int*)d_in[0];       // [B*S]
    const uint8_t*  packed   = (const uint8_t*)d_in[1];   // [V, D/2]
    const float*    absmax   = (const float*)d_in[2];     // [V]
    const float*    codebook = (const float*)d_in[3];     // [16]
    float*          out      = (float*)d_out;             // [B*S, D] f32

    const int n_tokens = in_sizes[0];                     // B*S = 32768
    const int D        = out_size / n_tokens;             // 1024
    const int halfD    = D >> 1;                          // 512
    const int tpt      = D >> 4;                          // threads/token = 64
    // tpt is a power of two (>=32) for this reference (D = 1024).
    int tok_shift = 0;
    while ((1 << tok_shift) < tpt) ++tok_shift;           // 6
    const int tok_mask = tpt - 1;

    const int total_threads = n_tokens * tpt;             // 2,097,152
    const int block = 256;
    const int grid  = (total_threads + block - 1) / block;

    nf4_embed_kernel<<<grid, block, 0, stream>>>(
        x, packed, absmax, codebook, out,
        total_threads, tok_shift, tok_mask, halfD, D);
}